// LearnedFrequencyFilterBank_80728205296210
// MI455X (gfx1250) — compile-verified
//
#include <hip/hip_runtime.h>

#define BATCH   8
#define SEQ     2048
#define HID     1024
#define NELEM   (SEQ * HID)        // 2,097,152 per batch
#define RANK_K  (NELEM / 2)        // 0-indexed ascending rank of kth value (N - k)
#define HIST_BLOCKS 512
#define NVEC    (NELEM / 4)        // float4 count per batch: 524,288
#define VPB     (NVEC / HIST_BLOCKS) // 1024 vec4 per block

typedef __attribute__((ext_vector_type(4))) int v4i;

__device__ __forceinline__ float sigmoidf_fast(float x) {
    // v_exp_f32 + v_add + v_rcp_f32 (avoid the IEEE divide expansion)
    return __builtin_amdgcn_rcpf(1.0f + __expf(-x));
}

__device__ __forceinline__ int band_of(int t) {
    // approx_len = 128; band 0 = [0,128); band b = [128*2^(b-1), 128*2^b)
    return (t < 128) ? 0 : (32 - __clz(t >> 7));
}

// ---- CDNA5 async global->LDS copy (ASYNCcnt-tracked), with safe fallback ----
__device__ __forceinline__ void async_copy16(void* lds_dst, const void* gsrc) {
#if __has_builtin(__builtin_amdgcn_global_load_async_to_lds_b128)
    __builtin_amdgcn_global_load_async_to_lds_b128(
        (__attribute__((address_space(1))) v4i*)gsrc,
        (__attribute__((address_space(3))) v4i*)lds_dst,
        0, 0);
#else
    *(float4*)lds_dst = *(const float4*)gsrc;
#endif
}

__device__ __forceinline__ void async_wait0() {
#if __has_builtin(__builtin_amdgcn_s_wait_asynccnt)
    __builtin_amdgcn_s_wait_asynccnt(0);
#elif __has_builtin(__builtin_amdgcn_global_load_async_to_lds_b128)
    asm volatile("s_wait_asynccnt 0" ::: "memory");
#endif
}

// -------------------------------------------------------------------------
// Init: sigmoid tables, zero histograms, per-batch select state
// -------------------------------------------------------------------------
__global__ __launch_bounds__(256) void ffb_init(const float* __restrict__ band_imp,
                                                const float* __restrict__ dim_imp,
                                                float* __restrict__ sb,
                                                float* __restrict__ sd,
                                                unsigned* __restrict__ hist,
                                                unsigned* __restrict__ state) {
    int i = blockIdx.x * 256 + threadIdx.x;
    if (i < 5 * HID) sb[i] = sigmoidf_fast(band_imp[i]);
    int j = i - 5 * HID;
    if (j >= 0 && j < HID) sd[j] = sigmoidf_fast(dim_imp[j]);
    int h = i - 6 * HID;
    if (h >= 0 && h < BATCH * 256) hist[h] = 0u;
    int s = i - 6 * HID - BATCH * 256;
    if (s >= 0 && s < 2 * BATCH) state[s] = (s & 1) ? (unsigned)RANK_K : 0u;
}

// -------------------------------------------------------------------------
// Radix-select histogram pass: 256-bin LDS histogram over one byte of the
// float bit pattern of ci = sb[band,d]*sd[d]*|coeff|, filtered by prefix.
// Sigmoid tables staged into LDS via async global->LDS (ASYNCcnt).
// -------------------------------------------------------------------------
__global__ __launch_bounds__(256) void ffb_hist(const float* __restrict__ coeffs,
                                                const float* __restrict__ sb,
                                                const float* __restrict__ sd,
                                                unsigned* __restrict__ hist,
                                                const unsigned* __restrict__ state,
                                                int pass) {
    __shared__ float    s_sb[5 * HID];   // 20 KB
    __shared__ float    s_sd[HID];       // 4 KB
    __shared__ unsigned s_hist[256];     // 1 KB
    const int tid = threadIdx.x;
    const int b   = blockIdx.y;

    // Bulk-stage 24 KB of sigmoid tables: 6 async b128 ops per lane
#pragma unroll
    for (int i = 0; i < 5; ++i) {
        const int o = i * 1024 + tid * 4;
        async_copy16(&s_sb[o], &sb[o]);
    }
    async_copy16(&s_sd[tid * 4], &sd[tid * 4]);
    s_hist[tid] = 0u;
    async_wait0();
    __syncthreads();

    const unsigned prefix = state[2 * b];
    const int shift  = 24 - 8 * pass;
    const int pshift = 32 - 8 * pass;   // only evaluated when pass > 0

    const float4* src = (const float4*)(coeffs + (size_t)b * NELEM);
    const int base = blockIdx.x * VPB;

    for (int j = tid; j < VPB; j += 256) {
        const int v = base + j;
        if (j + 256 < VPB) __builtin_prefetch((const void*)(src + v + 256), 0, 1);
        const float4 c = src[v];
        const int e = v << 2;
        const int d = e & (HID - 1);
        const int t = e >> 10;
        const int band = band_of(t);
        const float* sbp = s_sb + band * HID + d;
        const float* sdp = s_sd + d;
        float ci[4];
        ci[0] = fabsf(c.x) * sbp[0] * sdp[0];
        ci[1] = fabsf(c.y) * sbp[1] * sdp[1];
        ci[2] = fabsf(c.z) * sbp[2] * sdp[2];
        ci[3] = fabsf(c.w) * sbp[3] * sdp[3];
#pragma unroll
        for (int q = 0; q < 4; ++q) {
            const unsigned bits = __float_as_uint(ci[q]);
            const bool match = (pass == 0) || ((bits >> pshift) == prefix);
            if (match) atomicAdd(&s_hist[(bits >> shift) & 0xFFu], 1u);
        }
    }
    __syncthreads();
    const unsigned cnt = s_hist[tid];
    if (cnt) atomicAdd(&hist[b * 256 + tid], cnt);
}

// -------------------------------------------------------------------------
// Scan pass: pick the bin containing the rank, refine prefix, reset hist.
// -------------------------------------------------------------------------
__global__ __launch_bounds__(256) void ffb_scan(unsigned* __restrict__ hist,
                                                unsigned* __restrict__ state,
                                                float* __restrict__ thr,
                                                int pass) {
    __shared__ unsigned sh[256];
    const int b   = blockIdx.x;
    const int tid = threadIdx.x;
    sh[tid] = hist[b * 256 + tid];
    hist[b * 256 + tid] = 0u;     // ready for next pass
    __syncthreads();
    if (tid == 0) {
        unsigned rank   = state[2 * b + 1];
        unsigned prefix = state[2 * b];
        unsigned cum = 0u, chosen = 255u;
        for (int i = 0; i < 256; ++i) {
            const unsigned c = sh[i];
            if (cum + c > rank) { chosen = (unsigned)i; break; }
            cum += c;
        }
        const unsigned np = (prefix << 8) | chosen;
        state[2 * b]     = np;
        state[2 * b + 1] = rank - cum;
        if (pass == 3) thr[b] = __uint_as_float(np);
    }
}

// -------------------------------------------------------------------------
// Epilogue: filtered = c * sigmoid((imp*|c| - thr)/|temp|); write 3 outputs
// -------------------------------------------------------------------------
__global__ __launch_bounds__(256) void ffb_finalize(const float* __restrict__ coeffs,
                                                    const float* __restrict__ sb,
                                                    const float* __restrict__ sd,
                                                    const float* __restrict__ thr,
                                                    const float* __restrict__ temperature,
                                                    float* __restrict__ out) {
    const long long idx = (long long)blockIdx.x * 256 + threadIdx.x; // vec4 index
    const long long e   = idx << 2;
    const int b = (int)(e >> 21);               // / NELEM
    const int r = (int)(e & (NELEM - 1));
    const int t = r >> 10;
    const int d = r & (HID - 1);
    const int band = band_of(t);

    const float4 c  = ((const float4*)coeffs)[idx];
    const float  tb = thr[b];
    const float  invT = __builtin_amdgcn_rcpf(fabsf(temperature[0]));

    const float* sbp = sb + band * HID + d;
    const float* sdp = sd + d;

    float cc[4] = {c.x, c.y, c.z, c.w};
    float ff[4], mm[4], ii[4];
#pragma unroll
    for (int q = 0; q < 4; ++q) {
        const float imp = sbp[q] * sdp[q];
        const float ci  = imp * fabsf(cc[q]);
        const float mk  = sigmoidf_fast((ci - tb) * invT);
        ii[q] = imp;
        mm[q] = mk;
        ff[q] = cc[q] * mk;
    }

    float4* outF = (float4*)out;
    float4* outM = outF + (long long)BATCH * NVEC;
    float4* outI = outM + (long long)BATCH * NVEC;
    outF[idx] = make_float4(ff[0], ff[1], ff[2], ff[3]);
    outM[idx] = make_float4(mm[0], mm[1], mm[2], mm[3]);
    outI[idx] = make_float4(ii[0], ii[1], ii[2], ii[3]);
}

// -------------------------------------------------------------------------
extern "C" void kernel_launch(void* const* d_in, const int* in_sizes, int n_in,
                              void* d_out, int out_size, void* d_ws, size_t ws_size,
                              hipStream_t stream) {
    const float* coeffs      = (const float*)d_in[0];
    const float* band_imp    = (const float*)d_in[1];
    const float* dim_imp     = (const float*)d_in[2];
    const float* temperature = (const float*)d_in[3];

    // workspace layout (all < 33 KB)
    float*    sb    = (float*)d_ws;               // 5*1024 f32
    float*    sd    = sb + 5 * HID;               // 1024 f32
    unsigned* hist  = (unsigned*)(sd + HID);      // BATCH*256 u32
    unsigned* state = hist + BATCH * 256;         // BATCH*2 u32 (prefix, rank)
    float*    thr   = (float*)(state + 2 * BATCH); // BATCH f32

    ffb_init<<<33, 256, 0, stream>>>(band_imp, dim_imp, sb, sd, hist, state);

    for (int pass = 0; pass < 4; ++pass) {
        ffb_hist<<<dim3(HIST_BLOCKS, BATCH), 256, 0, stream>>>(coeffs, sb, sd, hist, state, pass);
        ffb_scan<<<BATCH, 256, 0, stream>>>(hist, state, thr, pass);
    }

    const int fin_blocks = (int)(((long long)BATCH * NVEC) / 256); // 16384
    ffb_finalize<<<fin_blocks, 256, 0, stream>>>(coeffs, sb, sd, thr, temperature, (float*)d_out);
}